// SPGG_Fermi_29274497090091
// MI455X (gfx1250) — compile-verified
//
#include <hip/hip_runtime.h>

// ---------------------------------------------------------------------------
// SPGG Fermi update, 4096x4096 periodic lattice.
//   profit = (R/5) * N2(t) - 5*t          (N2 = plus-stencil applied twice,
//                                          13-point closed form; c_matrix==t)
//   W      = 1/(1+exp((P - P_nbr)/K)),  K=0.1  -> *10
//   out    = (lp <= W) ? t_nbr : t      for the one direction in ld
// Bandwidth-bound: ~268 MB minimum -> ~11.5us floor @ 23.3 TB/s.
// CDNA5 paths: TDM tensor_load_to_lds stages interior tiles into LDS async
// (TENSORcnt); NT temporal hints keep the reused t/profit arrays (128 MB)
// resident in the 192 MB L2 while streaming lp/ld/out past it.
// ---------------------------------------------------------------------------

#define L       4096
#define LMASK   4095
#define TILE_W  128
#define TILE_H  32
#define HALO    2
#define FRAME_W (TILE_W + 2*HALO)   // 132
#define FRAME_H (TILE_H + 2*HALO)   // 36
#define PITCH   FRAME_W             // LDS row pitch in floats

#ifndef __has_builtin
#define __has_builtin(x) 0
#endif

#if __has_builtin(__builtin_amdgcn_tensor_load_to_lds)
#define HAVE_TDM 1
#else
#define HAVE_TDM 0
#endif

typedef unsigned int        u32;
typedef unsigned long long  u64;
typedef __attribute__((ext_vector_type(4))) u32   v4u32;
typedef __attribute__((ext_vector_type(4))) int   v4i32;
typedef __attribute__((ext_vector_type(8))) int   v8i32;
typedef __attribute__((ext_vector_type(4))) float f32x4;
typedef __attribute__((ext_vector_type(4))) int   i32x4;

typedef __attribute__((address_space(3))) float lds_float_t;

// ---------------------------------------------------------------------------
// Kernel 1: profit = 0.6 * N2(t) - 5*t  over a 128x32 tile (+halo 2).
// Interior staged by the Tensor Data Mover (async, LDS-pitch padding via
// pad_interval/pad_amount); periodic-wrap halo filled concurrently by
// normal global loads.
// ---------------------------------------------------------------------------
__global__ __launch_bounds__(256)
void spgg_profit_kernel(const float* __restrict__ t, float* __restrict__ profit) {
  __shared__ float tile[FRAME_H * PITCH];

  const int tid = threadIdx.x;
  const int c0  = blockIdx.x * TILE_W;   // tile origin (global col)
  const int r0  = blockIdx.y * TILE_H;   // tile origin (global row)

#if HAVE_TDM
  if (tid < 32) {
    // LDS byte address of the interior region (frame row 2, col 2).
    u32 lds_base = (u32)(u64)(lds_float_t*)tile;
    u32 lds_int  = lds_base + (u32)((HALO * PITCH + HALO) * 4);
    u64 ga = (u64)t + ((((u64)r0 * (u64)L) + (u64)c0) << 2);

    // D# Group 0: count=1 | lds_addr | global_addr | type=2
    v4u32 g0 = { 1u,
                 lds_int,
                 (u32)ga,
                 (u32)((ga >> 32) & 0x1FFFFFFu) | (2u << 30) };
    // D# Group 1:
    //  w0: data_size=2 (4B), pad_enable=1, pad_interval=6 (128 DW), pad_amount=3 (4 DW)
    //  w1: tensor_dim0[15:0]=4096 in bits[31:16]
    //  w2: tensor_dim0[31:16]=0 | tensor_dim1[15:0]=4096
    //  w3: tensor_dim1[31:16]=0 | tile_dim0=128
    //  w4: tile_dim1=32 | tile_dim2=0
    //  w5: tensor_dim0_stride[31:0]=4096 ; w6/w7: hi bits / dim1 stride = 0
    v8i32 g1 = { (int)((2u << 16) | (1u << 20) | (6u << 22) | (3u << 25)),
                 (int)((u32)L << 16),
                 (int)((u32)L << 16),
                 (int)((u32)TILE_W << 16),
                 (int)TILE_H,
                 (int)L,
                 0, 0 };
    v4i32 g2 = {0, 0, 0, 0};   // 2D tensor: groups 2/3 unused
    v4i32 g3 = {0, 0, 0, 0};
#if defined(__clang_major__) && (__clang_major__ >= 23)
    v8i32 g4 = {0, 0, 0, 0, 0, 0, 0, 0};
    __builtin_amdgcn_tensor_load_to_lds(g0, g1, g2, g3, g4, 0);
#else
    __builtin_amdgcn_tensor_load_to_lds(g0, g1, g2, g3, 0);
#endif
  }
#endif

  // Periodic-wrap halo (and, without TDM, the interior too) via global loads.
  for (int i = tid; i < FRAME_H * FRAME_W; i += 256) {
    const int ly = i / FRAME_W;
    const int lx = i - ly * FRAME_W;
#if HAVE_TDM
    const bool need = (ly < HALO) | (ly >= TILE_H + HALO) |
                      (lx < HALO) | (lx >= TILE_W + HALO);
#else
    const bool need = true;
#endif
    if (need) {
      const int gy = (r0 + ly - HALO) & LMASK;
      const int gx = (c0 + lx - HALO) & LMASK;
      tile[ly * PITCH + lx] = t[(size_t)gy * L + gx];
    }
  }

#if HAVE_TDM
  if (tid < 32) {
#if __has_builtin(__builtin_amdgcn_s_wait_tensorcnt)
    __builtin_amdgcn_s_wait_tensorcnt(0);
#else
    asm volatile("s_wait_tensorcnt 0x0" ::: "memory");
#endif
  }
#endif
  __syncthreads();

  // Each thread: one column, 16 rows, register sliding window over 5 rows.
  const int lx    = HALO + (tid & (TILE_W - 1));
  const int ybase = HALO + ((tid >> 7) << 4);     // frame row 2 or 18

#define TT(r, c) tile[(r) * PITCH + (c)]
  float a_m2 = TT(ybase - 2, lx);
  float a_m1 = TT(ybase - 1, lx);
  float b_m1 = TT(ybase - 1, lx - 1) + TT(ybase - 1, lx + 1);
  float a_c  = TT(ybase    , lx);
  float b_c  = TT(ybase    , lx - 1) + TT(ybase    , lx + 1);
  float a_p1 = TT(ybase + 1, lx);
  float b_p1 = TT(ybase + 1, lx - 1) + TT(ybase + 1, lx + 1);

#pragma unroll
  for (int j = 0; j < 16; ++j) {
    const int r = ybase + j;                       // center frame row
    const float a_p2 = TT(r + 2, lx);
    const float b_p2 = TT(r + 2, lx - 1) + TT(r + 2, lx + 1);
    const float e0   = TT(r, lx - 2) + TT(r, lx + 2);
    // 13-point double-plus stencil: w = {5 ctr, 2 axial1, 2 diag, 1 axial2}
    const float n2 = 5.0f * a_c + 2.0f * b_c + e0
                   + 2.0f * ((a_m1 + a_p1) + (b_m1 + b_p1))
                   + (a_m2 + a_p2);
    const float prof = 0.6f * n2 - 5.0f * a_c;     // (R/5)*N2 - 5*c, R=3
    const int gy = r0 + (r - HALO);
    const int gx = c0 + (lx - HALO);
    profit[(size_t)gy * L + gx] = prof;
    // shift window
    a_m2 = a_m1;
    a_m1 = a_c;  b_m1 = b_c;
    a_c  = a_p1; b_c  = b_p1;
    a_p1 = a_p2; b_p1 = b_p2;
  }
#undef TT
}

// ---------------------------------------------------------------------------
// Kernel 2: Fermi adoption, 4 cells per thread.
//   - center streams (t, profit, lp, ld) loaded as b128
//   - lp/ld are single-use -> non-temporal loads (TH=NT, spare L2 for reuse)
//   - out is single-use    -> non-temporal store
//   - neighbor gathers of profit/t hit L2 (t+profit = 128 MB < 192 MB L2)
// ---------------------------------------------------------------------------
__global__ __launch_bounds__(256)
void spgg_fermi_kernel(const float* __restrict__ t,
                       const float* __restrict__ lp,
                       const int*   __restrict__ ld,
                       const float* __restrict__ profit,
                       float* __restrict__ out) {
  const int x0 = (blockIdx.x * 256 + threadIdx.x) << 2;   // 4 cells / thread
  const int y  = blockIdx.y;
  const size_t i = (size_t)y * L + (size_t)x0;

  const f32x4 tc = *(const f32x4*)(t + i);
  const f32x4 P  = *(const f32x4*)(profit + i);
  const f32x4 p  = __builtin_nontemporal_load((const f32x4*)(lp + i));
  const i32x4 k4 = __builtin_nontemporal_load((const i32x4*)(ld + i));

  f32x4 res;
#pragma unroll
  for (int e = 0; e < 4; ++e) {
    const int x = x0 + e;
    const int k = k4[e];
    // k: 0 -> (y, x-1), 1 -> (y, x+1), 2 -> (y-1, x), 3 -> (y+1, x)
    const int dx = (k == 0) ? -1 : ((k == 1) ? 1 : 0);
    const int dy = (k == 2) ? -1 : ((k == 3) ? 1 : 0);
    const size_t ni = (size_t)((y + dy) & LMASK) * L + (size_t)((x + dx) & LMASK);

    const float Pn = profit[ni];
    const float tn = t[ni];

    const float W = 1.0f / (1.0f + expf((P[e] - Pn) * 10.0f));  // K = 0.1
    res[e] = (p[e] <= W) ? tn : tc[e];
  }
  __builtin_nontemporal_store(res, (f32x4*)(out + i));
}

// ---------------------------------------------------------------------------
extern "C" void kernel_launch(void* const* d_in, const int* in_sizes, int n_in,
                              void* d_out, int out_size, void* d_ws, size_t ws_size,
                              hipStream_t stream) {
  const float* t  = (const float*)d_in[0];   // type_t_matrix, float32 {0,1}
  const float* lp = (const float*)d_in[1];   // learning_probabilities
  const int*   ld = (const int*)d_in[2];     // learning_direction, int32 [0,4)
  float* profit = (float*)d_ws;              // 64 MB scratch
  float* out    = (float*)d_out;

  dim3 blk(256);
  dim3 g1(L / TILE_W, L / TILE_H);           // 32 x 128 tiles
  hipLaunchKernelGGL(spgg_profit_kernel, g1, blk, 0, stream, t, profit);

  dim3 g2(L / (256 * 4), L);                 // 4 x 4096, 4 cells/thread
  hipLaunchKernelGGL(spgg_fermi_kernel, g2, blk, 0, stream, t, lp, ld, profit, out);
}